// UltimateLiftSplatShoot_19576460935653
// MI455X (gfx1250) — compile-verified
//
#include <hip/hip_runtime.h>

// ---------------------------------------------------------------------------
// Lift-Splat-Shoot voxel pooling for MI455X (gfx1250, wave32).
//   geometry:  V_WMMA_F32_16X16X4_F32  (batched 3x4 affine, 16 points/op)
//   scatter :  global_atomic_add_f32 into L2-resident 41MB output
// ---------------------------------------------------------------------------

typedef __attribute__((ext_vector_type(2))) float v2f;
typedef __attribute__((ext_vector_type(8))) float v8f;

#define B_   4
#define N_   6
#define D_   59
#define H_   16
#define W_   44
#define C_   64
#define HW_   (H_ * W_)          // 704
#define DHW_  (D_ * HW_)         // 41536  (== 16 * 2596 exactly)
#define GROUPS_  (DHW_ / 16)     // 2596
#define WAVES_PER_BLOCK_ 8
#define NX0_ 200
#define NX1_ 200
#define CH_STRIDE_ (NX0_ * NX1_) // 40000 floats between channels in output
#define OUT_ELEMS_ (B_ * C_ * NX0_ * NX1_)  // 10,240,000 floats

// ---------------------------------------------------------------------------
// 3x3 inverse via adjugate with det regularized by 1e-8 (matches reference).
// ---------------------------------------------------------------------------
__device__ __forceinline__ void inv3(const float* m, float* inv) {
  float a11 = m[0], a12 = m[1], a13 = m[2];
  float a21 = m[3], a22 = m[4], a23 = m[5];
  float a31 = m[6], a32 = m[7], a33 = m[8];
  float det = a11 * a22 * a33 + a12 * a23 * a31 + a13 * a21 * a32
            - a13 * a22 * a31 - a12 * a21 * a33 - a11 * a23 * a32 + 1e-8f;
  inv[0] = (a22 * a33 - a23 * a32) / det;
  inv[1] = (a13 * a32 - a12 * a33) / det;
  inv[2] = (a12 * a23 - a13 * a22) / det;
  inv[3] = (a23 * a31 - a21 * a33) / det;
  inv[4] = (a11 * a33 - a13 * a31) / det;
  inv[5] = (a13 * a21 - a11 * a23) / det;
  inv[6] = (a21 * a32 - a22 * a31) / det;
  inv[7] = (a12 * a31 - a11 * a32) / det;
  inv[8] = (a11 * a22 - a12 * a21) / det;
}

// ---------------------------------------------------------------------------
// Zero-fill output (harness poisons d_out with 0xAA).
// ---------------------------------------------------------------------------
__global__ void lss_zero(float4* __restrict__ out) {
  int i = blockIdx.x * blockDim.x + threadIdx.x;
  if (i < OUT_ELEMS_ / 4) out[i] = make_float4(0.f, 0.f, 0.f, 0.f);
}

// ---------------------------------------------------------------------------
// Per-(b,n) preprocessing: build two 3x4 affine matrices into d_ws:
//   mat1 = [ inv(post_rots) | -inv(post_rots) @ post_trans ]
//   mat2 = [ rots @ inv(intrins) | trans ]
// Layout: mats[bn*24 + row*4 + col], mat2 at offset 12.
// ---------------------------------------------------------------------------
__global__ void lss_prep(const float* __restrict__ rots,
                         const float* __restrict__ trans,
                         const float* __restrict__ intrins,
                         const float* __restrict__ post_rots,
                         const float* __restrict__ post_trans,
                         float* __restrict__ mats) {
  int bn = threadIdx.x;
  if (bn >= B_ * N_) return;
  float pr[9], ipr[9], K[9], iK[9], R[9];
  for (int i = 0; i < 9; ++i) {
    pr[i] = post_rots[bn * 9 + i];
    K[i]  = intrins[bn * 9 + i];
    R[i]  = rots[bn * 9 + i];
  }
  inv3(pr, ipr);
  inv3(K, iK);
  float pt0 = post_trans[bn * 3 + 0];
  float pt1 = post_trans[bn * 3 + 1];
  float pt2 = post_trans[bn * 3 + 2];
  float* m = mats + bn * 24;
  for (int i = 0; i < 3; ++i) {
    // mat1 row i
    m[i * 4 + 0] = ipr[i * 3 + 0];
    m[i * 4 + 1] = ipr[i * 3 + 1];
    m[i * 4 + 2] = ipr[i * 3 + 2];
    m[i * 4 + 3] = -(ipr[i * 3 + 0] * pt0 + ipr[i * 3 + 1] * pt1 + ipr[i * 3 + 2] * pt2);
    // mat2 row i = (rots @ inv(intrins)) row i, plus trans
    float c0 = R[i * 3 + 0] * iK[0] + R[i * 3 + 1] * iK[3] + R[i * 3 + 2] * iK[6];
    float c1 = R[i * 3 + 0] * iK[1] + R[i * 3 + 1] * iK[4] + R[i * 3 + 2] * iK[7];
    float c2 = R[i * 3 + 0] * iK[2] + R[i * 3 + 1] * iK[5] + R[i * 3 + 2] * iK[8];
    m[12 + i * 4 + 0] = c0;
    m[12 + i * 4 + 1] = c1;
    m[12 + i * 4 + 2] = c2;
    m[12 + i * 4 + 3] = trans[bn * 3 + i];
  }
}

// ---------------------------------------------------------------------------
// Main kernel. grid = (ceil(GROUPS/8), B*N), block = 256 (8 waves).
// Each wave: 16 points through two WMMA 16x16x4 f32 affine transforms,
// then 16 x (64-channel) atomic scatter with each lane owning 2 channels.
//
// WMMA f32 16x16x4 operand layout (ISA 7.12.2):
//   A (16x4): lane L   holds A[L][0] (v0), A[L][1] (v1)
//             lane L+16 holds A[L][2] (v0), A[L][3] (v1)
//   B (4x16): lane n   holds B[0][n] (v0), B[1][n] (v1)
//             lane n+16 holds B[2][n] (v0), B[3][n] (v1)
//   D (16x16): VGPR r, lanes 0-15 -> D[r][lane]   (rows 0..7)
// Lane pair (L, L+16) both evaluate point L, so B needs no shuffles.
// ---------------------------------------------------------------------------
__global__ void __launch_bounds__(256)
lss_main(const float* __restrict__ x,
         const float* __restrict__ frustum,
         const float* __restrict__ mats,
         float* __restrict__ out) {
  const int lane = threadIdx.x & 31;
  const int wave = threadIdx.x >> 5;
  const int g = blockIdx.x * WAVES_PER_BLOCK_ + wave;   // point group (16 pts)
  if (g >= GROUPS_) return;                              // uniform per wave

  const int bn = blockIdx.y;
  const int b  = bn / N_;
  const int half = lane & 15;
  const int hi   = lane >> 4;

  // --- A operands: two 3x4 affine matrices (rows 3..15 are zero) ----------
  const float* m = mats + bn * 24;
  v2f a1 = {0.f, 0.f}, a2 = {0.f, 0.f};
  if (half < 3) {
    a1.x = m[half * 4 + hi * 2 + 0];
    a1.y = m[half * 4 + hi * 2 + 1];
    a2.x = m[12 + half * 4 + hi * 2 + 0];
    a2.y = m[12 + half * 4 + hi * 2 + 1];
  }

  // --- B operand #1: homogeneous frustum points ---------------------------
  const int p = g * 16 + half;                  // point index within (b,n)
  const float* f = frustum + (size_t)p * 3;     // (D,H,W,3): [x_pix, y_pix, depth]
  float fx = f[0], fy = f[1], fz = f[2];
  v2f b1;
  b1.x = hi ? fz : fx;
  b1.y = hi ? 1.0f : fy;

  v8f cz = {};
  // q = inv(post_rots) @ (frustum - post_trans)
  v8f c1 = __builtin_amdgcn_wmma_f32_16x16x4_f32(
      false, a1, false, b1, (short)0, cz, false, false);

  // --- nonlinear: r = (qx*qz, qy*qz, qz); broadcast qz to lane+16 ---------
  float qz = __shfl(c1[2], half, 32);
  v2f b2;
  b2.x = hi ? qz : c1[0] * qz;
  b2.y = hi ? 1.0f : c1[1] * qz;

  // s = (rots @ inv(intrins)) @ r + trans
  v8f c2 = __builtin_amdgcn_wmma_f32_16x16x4_f32(
      false, a2, false, b2, (short)0, cz, false, false);

  // --- voxelize (lanes 0..15 hold valid rows 0..2) ------------------------
  // geom = trunc((pts - (BX - DX/2)) / DX); BX-DX/2 = (-50,-50,-10)
  float sx = c2[0], sy = c2[1], sz = c2[2];
  int gx = (int)((sx + 50.0f) / 0.5f);
  int gy = (int)((sy + 50.0f) / 0.5f);
  int gz = (int)((sz + 10.0f) / 20.0f);
  // int-cast truncation spuriously maps (-1,0) -> 0; the reference does the
  // same truncation, so keep it bit-identical.
  int valid = (hi == 0) & (gx >= 0) & (gx < NX0_) & (gy >= 0) & (gy < NX1_) &
              (sx + 50.0f >= 0.0f ? 1 : (gx != 0)) * 0 + // no-op; see note above
              ((gz == 0) & (sz + 10.0f >= 0.0f || true));
  valid = (hi == 0) && gx >= 0 && gx < NX0_ && gy >= 0 && gy < NX1_ && gz == 0;

  // output base for channel 0: out[b][c][gx][gy], channel stride = 40000
  int obase = b * (C_ * CH_STRIDE_) + gx * NX1_ + gy;

  // --- scatter: 16 points x 64 channels, 2 channels per lane --------------
  const size_t featRow = ((size_t)bn * DHW_ + (size_t)g * 16) * C_;
  #pragma unroll 4
  for (int j = 0; j < 16; ++j) {
    int v  = __shfl(valid, j, 32);   // wave-uniform after broadcast
    int ob = __shfl(obase, j, 32);
    if (v) {
      const float2 fv =
          *(const float2*)(x + featRow + (size_t)j * C_ + (size_t)lane * 2);
      if (j + 1 < 16) {
        __builtin_prefetch(x + featRow + (size_t)(j + 1) * C_ + (size_t)lane * 2,
                           0, 1);    // global_prefetch_b8
      }
      atomicAdd(out + ob + (lane * 2 + 0) * CH_STRIDE_, fv.x);
      atomicAdd(out + ob + (lane * 2 + 1) * CH_STRIDE_, fv.y);
    }
  }
}

// ---------------------------------------------------------------------------
extern "C" void kernel_launch(void* const* d_in, const int* in_sizes, int n_in,
                              void* d_out, int out_size, void* d_ws, size_t ws_size,
                              hipStream_t stream) {
  const float* x          = (const float*)d_in[0];
  const float* rots       = (const float*)d_in[1];
  const float* trans      = (const float*)d_in[2];
  const float* intrins    = (const float*)d_in[3];
  const float* post_rots  = (const float*)d_in[4];
  const float* post_trans = (const float*)d_in[5];
  const float* frustum    = (const float*)d_in[6];
  float* out  = (float*)d_out;
  float* mats = (float*)d_ws;   // 24 * 24 floats = 2304 bytes

  // 1) zero the (poisoned) output
  lss_zero<<<dim3(OUT_ELEMS_ / 4 / 256), dim3(256), 0, stream>>>((float4*)out);

  // 2) build per-camera affine matrices
  lss_prep<<<dim3(1), dim3(32), 0, stream>>>(rots, trans, intrins, post_rots,
                                             post_trans, mats);

  // 3) WMMA geometry + atomic scatter
  dim3 grid((GROUPS_ + WAVES_PER_BLOCK_ - 1) / WAVES_PER_BLOCK_, B_ * N_);
  lss_main<<<grid, dim3(256), 0, stream>>>(x, frustum, mats, out);
}